// PowerIterationPageRank_9835475108063
// MI455X (gfx1250) — compile-verified
//
#include <hip/hip_runtime.h>

// APPNP power iteration: x <- 0.85 * dropout(A) @ x + 0.15 * logits, 5 rounds.
// N = 100000 nodes, E = 1.6M edges, d = 48 (N%16==0, d==3*16 -> perfect WMMA tiling).

#define FDIM   48
#define ALPHA  0.15f
#define NPROP  5

typedef float v2f __attribute__((ext_vector_type(2)));
typedef float v8f __attribute__((ext_vector_type(8)));

// Counter-based RNG: pure function of (edge, iter) -> deterministic across replays.
__device__ __forceinline__ unsigned edge_hash(unsigned e, unsigned it) {
  unsigned h = e * 0x9E3779B1u ^ (it * 0x85EBCA77u + 0x165667B1u);
  h ^= h >> 15; h *= 0x2C1B3C6Du;
  h ^= h >> 12; h *= 0x297A2D39u;
  h ^= h >> 15;
  return h;
}

__global__ void zero_f4(float4* __restrict__ p, int n4) {
  int i = blockIdx.x * blockDim.x + threadIdx.x;
  if (i < n4) p[i] = make_float4(0.f, 0.f, 0.f, 0.f);
}

// One thread per (edge, float4 chunk); 12 chunks cover the 48-float row.
// Gather x[col] (L2-resident, 192B coalesced per edge), scale by inverted-dropout
// value, scatter-add into acc[row] with non-returning global_atomic_add_f32.
__global__ __launch_bounds__(256)
void spmm_dropout(const float* __restrict__ x,
                  const float* __restrict__ ev,
                  const int*   __restrict__ er,
                  const int*   __restrict__ ec,
                  float*       __restrict__ acc,
                  int E, int iter) {
  int tid = blockIdx.x * blockDim.x + threadIdx.x;
  int e = tid / 12;
  if (e >= E) return;
  int c = tid - e * 12;                    // float4 chunk id 0..11
  unsigned h = edge_hash((unsigned)e, (unsigned)iter);
  if (h & 0x80000000u) {                   // keep prob 0.5; whole edge group drops together
    float v = ev[e] * 2.0f;                // inverted dropout scale 1/keep
    int cn = ec[e];
    int rn = er[e];
    const float4 xv = *reinterpret_cast<const float4*>(x + (size_t)cn * FDIM + c * 4);
    float* dst = acc + (size_t)rn * FDIM + c * 4;
    atomicAdd(dst + 0, v * xv.x);
    atomicAdd(dst + 1, v * xv.y);
    atomicAdd(dst + 2, v * xv.z);
    atomicAdd(dst + 3, v * xv.w);
  }
}

// Per-wave 16x16 tile: D = sum_k (0.85*I_k)x(acc rows 4k..4k+3)
//                        + sum_k (0.15*I_k)x(logits rows 4k..4k+3)
// via 8x V_WMMA_F32_16X16X4_F32 (full fp32). Also zeroes acc for the next round.
// Layouts per CDNA5 ISA 7.12.2:
//   A 16x4 f32 : lane L holds M=L&15; VGPR p holds K = 2*(L>>4)+p
//   B  4x16 f32: lane L holds N=L&15; VGPR p holds K = 2*(L>>4)+p
//   C/D 16x16  : lane L holds N=L&15; VGPR k holds M = k + 8*(L>>4)
__global__ __launch_bounds__(192)
void combine_wmma(const float* __restrict__ acc,
                  const float* __restrict__ logits,
                  float*       __restrict__ xout,
                  float*       __restrict__ acc_clr,
                  int ntiles) {
  int w = (blockIdx.x * blockDim.x + threadIdx.x) >> 5;   // global wave id
  if (w >= ntiles) return;                                // uniform per-wave exit
  int lane = threadIdx.x & 31;
  int hh = lane >> 4;        // half-wave (0/1)
  int nl = lane & 15;        // lane within half
  int g  = w / 3;            // node group (16 rows)
  int f  = w - g * 3;        // feature tile (16 cols)
  int m0 = g << 4;
  int n0 = f << 4;
  int col = n0 + nl;

  // Per-lane B-operand elements: acc/logits rows r = 4k + 2*hh + p at this lane's column.
  float av[8], lv[8];
#pragma unroll
  for (int k = 0; k < 4; ++k)
#pragma unroll
    for (int p = 0; p < 2; ++p) {
      int r = 4 * k + 2 * hh + p;
      size_t idx = (size_t)(m0 + r) * FDIM + col;
      av[2 * k + p] = acc[idx];
      lv[2 * k + p] = logits[idx];
    }

  v8f d = {0.f, 0.f, 0.f, 0.f, 0.f, 0.f, 0.f, 0.f};
#pragma unroll
  for (int k = 0; k < 4; ++k) {
    v2f a, b;
    a.x = (nl == 4 * k + 2 * hh + 0) ? (1.0f - ALPHA) : 0.0f;  // 0.85 * identity slice
    a.y = (nl == 4 * k + 2 * hh + 1) ? (1.0f - ALPHA) : 0.0f;
    b.x = av[2 * k + 0];
    b.y = av[2 * k + 1];
    d = __builtin_amdgcn_wmma_f32_16x16x4_f32(false, a, false, b, (short)0, d, false, false);
  }
#pragma unroll
  for (int k = 0; k < 4; ++k) {
    v2f a, b;
    a.x = (nl == 4 * k + 2 * hh + 0) ? ALPHA : 0.0f;           // 0.15 * identity slice
    a.y = (nl == 4 * k + 2 * hh + 1) ? ALPHA : 0.0f;
    b.x = lv[2 * k + 0];
    b.y = lv[2 * k + 1];
    d = __builtin_amdgcn_wmma_f32_16x16x4_f32(false, a, false, b, (short)0, d, false, false);
  }

  // Store D tile (row M = k + 8*hh at this lane's column), 64B coalesced per row-half.
#pragma unroll
  for (int k = 0; k < 8; ++k)
    xout[(size_t)(m0 + k + 8 * hh) * FDIM + col] = d[k];

  // Fused re-zero of the accumulator for the next propagation round
  // (rows {4k+2hh, 4k+2hh+1} over both half-waves cover all 16 rows).
#pragma unroll
  for (int k = 0; k < 4; ++k)
#pragma unroll
    for (int p = 0; p < 2; ++p) {
      int r = 4 * k + 2 * hh + p;
      acc_clr[(size_t)(m0 + r) * FDIM + col] = 0.0f;
    }
}

extern "C" void kernel_launch(void* const* d_in, const int* in_sizes, int n_in,
                              void* d_out, int out_size, void* d_ws, size_t ws_size,
                              hipStream_t stream) {
  const float* logits = (const float*)d_in[0];
  const float* ev     = (const float*)d_in[1];
  const int*   er     = (const int*)d_in[2];
  const int*   ec     = (const int*)d_in[3];

  const int ND = in_sizes[0];        // N * 48
  const int E  = in_sizes[1];
  const int N  = ND / FDIM;

  float* xA  = (float*)d_ws;         // ping
  float* xB  = xA + ND;              // pong
  float* acc = xB + ND;              // SpMM accumulator

  // Zero the accumulator (ws is poisoned before timing; every call must be identical).
  int n4 = ND / 4;
  zero_f4<<<(n4 + 255) / 256, 256, 0, stream>>>((float4*)acc, n4);

  const int ntiles   = (N / 16) * 3;               // 16-row groups x 3 feature tiles
  const int cthreads = ntiles * 32;                // one wave32 per tile
  const int cblocks  = (cthreads + 191) / 192;     // 6 waves per block

  const int sthreads = E * 12;                     // (edge, float4-chunk) work items
  const int sblocks  = (sthreads + 255) / 256;

  const float* xin = logits;
  for (int it = 0; it < NPROP; ++it) {
    spmm_dropout<<<sblocks, 256, 0, stream>>>(xin, ev, er, ec, acc, E, it);
    float* xo = (it == NPROP - 1) ? (float*)d_out : ((it & 1) ? xB : xA);
    combine_wmma<<<cblocks, 192, 0, stream>>>(acc, logits, xo, acc, ntiles);
    xin = xo;
  }
}